// QCDAware_11605001634331
// MI455X (gfx1250) — compile-verified
//
#include <hip/hip_runtime.h>
#include <hip/hip_bf16.h>
#include <math.h>

#define EPS 1e-07f
#define BATCH 8
#define NPTS 2048
#define FEAT 6
#define LOG2E 1.4426950408889634f

typedef float v2f __attribute__((ext_vector_type(2)));
typedef float v8f __attribute__((ext_vector_type(8)));

// ---------------------------------------------------------------------------
// Kernel 1: per-batch preprocessing.
//   pow[n]  = exp(2*alpha*log(momentum[n]))        (momentum = emb[b,4,:])
//   d_min   = min_n pow[n] + EPS
//   c1l[b]  = log2(e) * beta^2 / (d_min * 1e4 * (radius^2 + EPS))
// Stored in workspace: ws_pow[B*N], ws_c1l[B].
// ---------------------------------------------------------------------------
__global__ __launch_bounds__(256) void qcd_prep_kernel(
    const float* __restrict__ emb,
    const float* __restrict__ alpha,
    const float* __restrict__ beta,
    const float* __restrict__ radius,
    float* __restrict__ ws_pow,
    float* __restrict__ ws_c1l)
{
    const int b   = blockIdx.x;
    const int tid = threadIdx.x;
    const float a2 = 2.0f * alpha[0];

    const float* mom = emb + ((size_t)b * FEAT + 4) * NPTS;
    float local_min = 3.402823466e+38f;

    for (int n = tid; n < NPTS; n += 256) {
        float m = mom[n];
        float p = expf(a2 * logf(m));          // accurate path, only 16K elems
        ws_pow[(size_t)b * NPTS + n] = p;
        local_min = fminf(local_min, p);
    }

    __shared__ float smin[256];
    smin[tid] = local_min;
    __syncthreads();
    for (int s = 128; s > 0; s >>= 1) {
        if (tid < s) smin[tid] = fminf(smin[tid], smin[tid + s]);
        __syncthreads();
    }

    if (tid == 0) {
        float dmin = smin[0] + EPS;
        float bt   = beta[0];
        float r    = radius[0];
        float sqr  = r * r + EPS;
        ws_c1l[b]  = LOG2E * (bt * bt) / (dmin * 10000.0f * sqr);
    }
}

// ---------------------------------------------------------------------------
// Kernel 2: one 16x16 output tile per wave via V_WMMA_F32_16X16X4_F32.
//   sqdist tile = A(16x4) x B(4x16) with
//     A row i = [-2x_i, -2y_i, |c_i|^2, 1]
//     B col j = [ x_j ,  y_j ,   1    , |c_j|^2]
//   then w_ij = 2^( s2l - sqdist * min(pow_i,pow_j) * c1l[b] )
//   where s2l = log2(e)*beta^2*1e-4 (uniform), c1l absorbs dmin/radius/log2e.
//   Epilogue per element: fmin + mul + fma(neg) + v_exp_f32 + store.
// ---------------------------------------------------------------------------
__global__ __launch_bounds__(256) void qcd_tile_kernel(
    const float* __restrict__ emb,
    const float* __restrict__ beta,
    const float* __restrict__ ws_pow,
    const float* __restrict__ ws_c1l,
    float* __restrict__ out)
{
    const int NT = NPTS / 16;                 // 128 tiles per dimension
    const int wave = threadIdx.x >> 5;        // 8 waves per block
    const int lane = threadIdx.x & 31;
    const int half = lane >> 4;               // 0 = lanes 0-15, 1 = lanes 16-31
    const int l15  = lane & 15;

    // Global tile id: tn fastest so the 8 waves of a block write 128
    // contiguous columns of the same 16-row strip.
    const long tile = (long)blockIdx.x * 8 + wave;
    const int  tn   = (int)(tile % NT);
    const int  tm   = (int)((tile / NT) % NT);
    const int  b    = (int)(tile / ((long)NT * NT));

    const float* cx = emb + ((size_t)b * FEAT + 1) * NPTS;  // coord x
    const float* cy = emb + ((size_t)b * FEAT + 2) * NPTS;  // coord y
    const float* pw = ws_pow + (size_t)b * NPTS;

    const int gi = tm * 16 + l15;   // row handled by this lane's A slots
    const int gj = tn * 16 + l15;   // col handled by this lane's B slots

    const float xi = cx[gi], yi = cy[gi];
    const float xj = cx[gj], yj = cy[gj];
    const float sqn_i = xi * xi + yi * yi;
    const float sqn_j = xj * xj + yj * yj;

    // A (16x4): VGPR0 holds K = 0 (lanes 0-15) / K = 2 (lanes 16-31)
    //           VGPR1 holds K = 1 (lanes 0-15) / K = 3 (lanes 16-31)
    v2f amat;
    amat.x = half ? sqn_i : (-2.0f * xi);
    amat.y = half ? 1.0f  : (-2.0f * yi);

    // B (4x16): VGPR0 holds row K=0 (lanes 0-15) / K=2 (lanes 16-31)
    //           VGPR1 holds row K=1 (lanes 0-15) / K=3 (lanes 16-31)
    v2f bmat;
    bmat.x = half ? 1.0f  : xj;
    bmat.y = half ? sqn_j : yj;

    v8f c = {};
    c = __builtin_amdgcn_wmma_f32_16x16x4_f32(
        /*neg_a=*/false, amat, /*neg_b=*/false, bmat,
        /*c_mod=*/(short)0, c, /*reuse_a=*/false, /*reuse_b=*/false);

    // Epilogue constants (log2-domain so v_exp_f32 is used raw)
    const float bt   = beta[0];
    const float s2l  = bt * bt * 1e-4f * LOG2E;  // log2e * beta^2 / 1e4
    const float c1l  = ws_c1l[b];                // log2e * beta^2/(dmin*1e4*sqr)
    const float pj   = pw[gj];                   // pow_momenta at lane's col
    const float prow = pw[gi];                   // pow_momenta at lane's row

    float* obase = out + ((size_t)b * NPTS) * NPTS;

    #pragma unroll
    for (int v = 0; v < 8; ++v) {
        const int ilocal = v + 8 * half;       // C/D: VGPR v -> rows v, v+8
        // Broadcast pow_momenta of row (tm*16 + ilocal): lane ilocal holds it.
        const float pi   = __shfl(prow, v + 8 * half, 32);
        const float minm = fminf(pi, pj);
        const float arg  = __builtin_fmaf(-(c[v] * minm), c1l, s2l);
        const float w    = __builtin_amdgcn_exp2f(arg);   // raw v_exp_f32
        obase[(size_t)(tm * 16 + ilocal) * NPTS + (tn * 16 + l15)] = w;
    }
}

extern "C" void kernel_launch(void* const* d_in, const int* in_sizes, int n_in,
                              void* d_out, int out_size, void* d_ws, size_t ws_size,
                              hipStream_t stream) {
    const float* emb    = (const float*)d_in[0];
    const float* alpha  = (const float*)d_in[1];
    const float* beta   = (const float*)d_in[2];
    const float* radius = (const float*)d_in[3];
    float* out = (float*)d_out;

    float* ws_pow = (float*)d_ws;              // B*N floats
    float* ws_c1l = ws_pow + BATCH * NPTS;     // B floats

    qcd_prep_kernel<<<BATCH, 256, 0, stream>>>(emb, alpha, beta, radius,
                                               ws_pow, ws_c1l);

    const int NT = NPTS / 16;                            // 128
    const long tiles = (long)BATCH * NT * NT;            // 131072
    const int blocks = (int)(tiles / 8);                 // 8 waves per block
    qcd_tile_kernel<<<blocks, 256, 0, stream>>>(emb, beta, ws_pow, ws_c1l, out);
}